// ScaleNumEmbed_25726854103624
// MI455X (gfx1250) — compile-verified
//
#include <hip/hip_runtime.h>
#include <cmath>

// ScaleNumEmbed for MI455X (gfx1250, wave32).
// out[t,d] = is_num[t] ? sum_n sigmoid(log(num[t])*W[n,d] + b[n,d]) : embeds[t,d]
//
// Bandwidth-bound: ~200 MB streamed => ~8.6 us floor at 23.3 TB/s.
// Rank-1 (feat x W + b) update done with V_WMMA_F32_16X16X4_F32 (only K=0 live),
// sigmoid via native v_exp_f32/v_rcp_f32, embeds/out streamed non-temporally.
//
// This revision replaces the final mask-select with an arithmetic blend
// (fma with a 0/1 mask) so the embeds load cannot be sunk under a divergent
// exec-mask branch: with random masks that branch never skips, it only added
// saveexec/branch/wait overhead around every one of the 8 row loads per tile.

typedef float v2f __attribute__((ext_vector_type(2)));
typedef float v8f __attribute__((ext_vector_type(8)));

#define D_DIM   768
#define N_LINS  4
#define WAVES_PER_BLOCK 4
#define DTILES_PER_WAVE 12   // 48 d-tiles of 16 / 4 waves

__global__ __launch_bounds__(128)
void scale_num_embed_wmma(const float* __restrict__ embeds,
                          const float* __restrict__ numbers,
                          const int*   __restrict__ isnum,
                          const float* __restrict__ Wm,
                          const float* __restrict__ Bias,
                          float*       __restrict__ out)
{
    const int lane  = threadIdx.x & 31;
    const int wave  = threadIdx.x >> 5;
    const int tbase = blockIdx.x * 16;        // 16 tokens per tile

    const int col = lane & 15;                // matrix column (d within tile)
    const int hi  = (lane >> 4) & 1;          // lanes 16-31 hold rows 8..15
    const float lo16 = (lane < 16) ? 1.0f : 0.0f;  // branchless lane select

    // ---- per-token feature in A-matrix layout -------------------------------
    // A is 16x4 f32: VGPR0 lanes0-15 = K=0 (live column = feat), all else 0.
    // numbers in [0.5, 100.5] -> native v_log_f32 * ln2 is safe & branchless.
    const int   tokA = tbase + col;
    const float nv   = numbers[tokA];              // unconditional, in-bounds
    const int   ia   = isnum[tokA];
    const float lnv  = __builtin_amdgcn_logf(nv) * 0.6931471805599453f;
    const float feat = ia ? lnv : 0.0f;            // v_cndmask

    v2f a;
    a.x = feat * lo16;                        // K=0 (lo lanes) / K=2 (hi)=0
    a.y = 0.0f;                               // K=1 / K=3 = 0

    // ---- per-row 0/1 blend masks (C/D layout: VGPR r, lane -> row r+8*hi) ---
    float m01[8];
    #pragma unroll
    for (int r = 0; r < 8; ++r)
        m01[r] = isnum[tbase + r + hi * 8] ? 1.0f : 0.0f;

    const float neg_log2e = -1.4426950408889634f;

    for (int t = 0; t < DTILES_PER_WAVE; ++t) {
        const int dbase = (wave * DTILES_PER_WAVE + t) * 16;

        float acc[8];
        #pragma unroll
        for (int r = 0; r < 8; ++r) acc[r] = 0.0f;

        #pragma unroll
        for (int n = 0; n < N_LINS; ++n) {
            // B is 4x16: VGPR0 lanes0-15 = row K=0 = W[n, dbase+col]; rest 0.
            const float wv = Wm[n * D_DIM + dbase + col];
            v2f bm;
            bm.x = wv * lo16;
            bm.y = 0.0f;

            // C = b[n, dbase+col] broadcast across all 16 rows (8 VGPRs).
            const float bv = Bias[n * D_DIM + dbase + col];
            v8f c = { bv, bv, bv, bv, bv, bv, bv, bv };

            // z[row, col] = feat[row] * W[n, col] + b[n, col]
            v8f z = __builtin_amdgcn_wmma_f32_16x16x4_f32(
                false, a, false, bm, (short)0, c, false, false);

            // sigmoid via native v_exp_f32 / v_rcp_f32, summed over n
            #pragma unroll
            for (int r = 0; r < 8; ++r) {
                float e = __builtin_amdgcn_exp2f(z[r] * neg_log2e); // exp(-z)
                acc[r] += __builtin_amdgcn_rcpf(1.0f + e);
            }
        }

        // ---- stream embeds in, arithmetic blend, stream out (NT both ways) --
        // ov = ev + m*(acc-ev): load is unconditionally demanded -> no
        // divergent saveexec/branch wrapper, fully coalesced NT stream.
        #pragma unroll
        for (int r = 0; r < 8; ++r) {
            const size_t idx = (size_t)(tbase + r + hi * 8) * D_DIM + dbase + col;
            const float  ev  = __builtin_nontemporal_load(&embeds[idx]);
            const float  ov  = __builtin_fmaf(m01[r], acc[r] - ev, ev);
            __builtin_nontemporal_store(ov, &out[idx]);
        }
    }
}

extern "C" void kernel_launch(void* const* d_in, const int* in_sizes, int n_in,
                              void* d_out, int out_size, void* d_ws, size_t ws_size,
                              hipStream_t stream) {
    const float* embeds  = (const float*)d_in[0];  // [B,S,D] f32
    const float* numbers = (const float*)d_in[1];  // [B,S]   f32
    const int*   isnum   = (const int*)  d_in[2];  // [B,S]   bool->int
    const float* Wm      = (const float*)d_in[3];  // [N,D]   f32
    const float* Bias    = (const float*)d_in[4];  // [N,D]   f32
    float*       out     = (float*)d_out;          // [B,S,D] f32

    const int tokens = in_sizes[1];                // B*S = 32768
    const int tiles  = tokens / 16;                // 2048 token-tiles

    scale_num_embed_wmma<<<tiles, 32 * WAVES_PER_BLOCK, 0, stream>>>(
        embeds, numbers, isnum, Wm, Bias, out);
}